// SoftPERLoss_5437428597068
// MI455X (gfx1250) — compile-verified
//
#include <hip/hip_runtime.h>
#include <stdint.h>

#define T_DIM 1024
#define B_DIM 64
#define C_DIM 512
#define UMAXC 256

typedef float v2f __attribute__((ext_vector_type(2)));
typedef float v8f __attribute__((ext_vector_type(8)));

// base-2 reformulation of softmin with tau = 0.2:
//   softmin(x..) = m - tau*ln2 * log2( sum 2^((m-x) * (1/tau)/ln2) )
#define K_SCALE 7.213475204444817f     // (1/tau)/ln2 = 5/ln2
#define TAU_LN2 0.13862943611198906f   // tau*ln2
#define LOG2E   1.4426950408889634f

__device__ __forceinline__ float exp2_fast(float x) { return __builtin_amdgcn_exp2f(x); }
__device__ __forceinline__ float log2_fast(float x) { return __builtin_amdgcn_logf(x); }

__device__ __forceinline__ float softmin2f(float a, float b) {
    float m = fminf(a, b);
    float s = exp2_fast((m - a) * K_SCALE) + exp2_fast((m - b) * K_SCALE);
    return m - TAU_LN2 * log2_fast(s);
}
__device__ __forceinline__ float softmin3f(float a, float b, float c) {
    float m = fminf(fminf(a, b), c);
    float s = exp2_fast((m - a) * K_SCALE) + exp2_fast((m - b) * K_SCALE)
            + exp2_fast((m - c) * K_SCALE);
    return m - TAU_LN2 * log2_fast(s);
}

// ---- CDNA5 async global->LDS copy (ASYNCcnt path) ----
__device__ __forceinline__ void async_copy_b64(unsigned lds_off, const void* gptr) {
    asm volatile("global_load_async_to_lds_b64 %0, %1, off"
                 :: "v"(lds_off), "v"(gptr) : "memory");
}
__device__ __forceinline__ void wait_async(int n) {  // n is block-uniform
    switch (n) {
    case 0:  asm volatile("s_wait_asynccnt 0x0" ::: "memory"); break;
    case 1:  asm volatile("s_wait_asynccnt 0x1" ::: "memory"); break;
    case 2:  asm volatile("s_wait_asynccnt 0x2" ::: "memory"); break;
    default: asm volatile("s_wait_asynccnt 0x3" ::: "memory"); break;
    }
}

// One workgroup per batch; 256 threads = 8 wave32.
// Thread i owns DP position u = i+1 of the (Umax+1)-row.
__global__ __launch_bounds__(256)
void softper_dp_kernel(const float* __restrict__ lp,     // (T,B,C)
                       const int*   __restrict__ targets,// (B*UMAX)
                       const int*   __restrict__ in_len, // (B)
                       const int*   __restrict__ tg_len, // (B)
                       float*       __restrict__ ws)     // (B) partial results
{
    const int b    = blockIdx.x;
    const int tid  = threadIdx.x;
    const int lane = tid & 31;
    const int wave = tid >> 5;

    __shared__ float rowbuf[4][C_DIM];   // 4-deep ring of prob rows (async target)
    __shared__ float prevA[UMAXC + 1];   // DP row (cost domain)
    __shared__ float Fs[UMAXC];          // exp-domain, block-rescaled f values
    __shared__ float scan_s[UMAXC];      // intra-block scan results (cost domain)
    __shared__ float bmin[16];           // per-block min (rescale reference)
    __shared__ float carry_l[16];        // inclusive cross-block scan
    __shared__ int   meta[1];            // flat target offset for this batch

    // ---- clamp lengths, compute this batch's flat-target offset ----
    if (tid == 0) {
        int o = 0;
        for (int k = 0; k < b; ++k) {
            int l = tg_len[k];
            l = l < 1 ? 1 : (l > UMAXC ? UMAXC : l);
            o += l;
        }
        meta[0] = o;
    }
    int Lb, Tb;
    {
        int l = tg_len[b]; Lb = l < 1 ? 1 : (l > UMAXC ? UMAXC : l);
        int t = in_len[b]; Tb = t < 1 ? 1 : (t > T_DIM ? T_DIM : t);
    }
    prevA[tid + 1] = (float)(tid + 1);   // i * INS_COST
    if (tid == 0) prevA[0] = 0.0f;
    __syncthreads();

    // unpacked target id for this thread's DP position (register, not LDS)
    int myTgt;
    {
        int gi = meta[0] + tid;
        const int gmax = B_DIM * UMAXC - 1;
        gi = gi > gmax ? gmax : gi;
        myTgt = (tid < Lb) ? targets[gi] : 0;
    }

    // ---- wave 0: precompute triangular-Toeplitz A (M[j][k] = r^(j-k), r=e^-5) ----
    float a0[4], a1[4];
    if (wave == 0) {
        const int col = lane & 15, hi = lane >> 4;
#pragma unroll
        for (int cc = 0; cc < 4; ++cc) {
            const int k0 = 4 * cc + 2 * hi;
            a0[cc] = (col >= k0)     ? exp2_fast(-K_SCALE * (float)(col - k0))     : 0.0f;
            a1[cc] = (col >= k0 + 1) ? exp2_fast(-K_SCALE * (float)(col - k0 - 1)) : 0.0f;
        }
    }

    // ---- bootstrap: async-copy rows 0..2 (8B per lane per row) ----
#pragma unroll
    for (int k = 0; k < 3; ++k) {
        if (k < Tb) {
            const float* rk = lp + ((size_t)k * B_DIM + b) * C_DIM;
            unsigned lo = (unsigned)(uintptr_t)(&rowbuf[k][tid * 2]);
            async_copy_b64(lo, (const void*)(rk + tid * 2));
        }
    }

    for (int t = 0; t < Tb; ++t) {
        const int cb = t & 3;
        // prefetch 3 rows ahead; wait until row t is resident
        if (t + 3 < Tb) {
            const float* rn = lp + ((size_t)(t + 3) * B_DIM + b) * C_DIM;
            unsigned ld = (unsigned)(uintptr_t)(&rowbuf[(t + 3) & 3][tid * 2]);
            async_copy_b64(ld, (const void*)(rn + tid * 2));
        }
        {
            int rem = Tb - 1 - t;
            rem = rem > 3 ? 3 : rem;
            wait_async(rem);
        }
        __syncthreads();                       // (1) publish row t

        // blank cost: -log(clip(exp(lp_blank), eps)) = min(-lp, -log(eps))
        const float bc   = fminf(-rowbuf[cb][0], 18.420680743952367f);
        const float cur0 = prevA[0] + bc;

        // f_u = softmin(prev[u]+bc, prev[u-1]+sc_u); invalid -> +INF
        float f = 1e30f;
        if (tid < Lb) {
            const float sc = 1.0f - exp2_fast(rowbuf[cb][myTgt] * LOG2E);
            f = softmin2f(prevA[tid + 1] + bc, prevA[tid] + sc);
        }
        // per-16-block min (half-wave reduction) for exp-domain rescale
        float m = f;
        m = fminf(m, __shfl_xor(m, 1, 16));
        m = fminf(m, __shfl_xor(m, 2, 16));
        m = fminf(m, __shfl_xor(m, 4, 16));
        m = fminf(m, __shfl_xor(m, 8, 16));
        Fs[tid] = exp2_fast((m - f) * K_SCALE);        // in (0,1]
        if ((tid & 15) == 0) bmin[tid >> 4] = m;
        __syncthreads();                       // (2) Fs/bmin visible to wave 0

        // ---- intra-block inclusive scan as triangular-Toeplitz WMMA matmul ----
        // S = M * F ; 16 blocks are the 16 N-columns; K=16 via 4 chained WMMAs.
        if (wave == 0) {
            const int col = lane & 15;
            const int hi  = lane >> 4;
            v8f acc = {0.f, 0.f, 0.f, 0.f, 0.f, 0.f, 0.f, 0.f};
#pragma unroll
            for (int cc = 0; cc < 4; ++cc) {
                const int k0 = 4 * cc + 2 * hi;
                v2f av, bv;
                av.x = a0[cc];
                av.y = a1[cc];
                bv.x = Fs[col * 16 + k0];
                bv.y = Fs[col * 16 + k0 + 1];
                acc = __builtin_amdgcn_wmma_f32_16x16x4_f32(
                          false, av, false, bv, (short)0, acc, false, false);
            }
            const float mc = bmin[col];
            float s15 = 0.0f;
#pragma unroll
            for (int v = 0; v < 8; ++v) {
                const int j = v + 8 * hi;      // C/D layout: VGPR v = rows v, v+8
                float sv = mc - TAU_LN2 * log2_fast(fmaxf(acc[v], 1e-37f));
                scan_s[col * 16 + j] = sv;
                if (v == 7) s15 = sv;          // j==15 lives on hi==1 lanes
            }
            // cross-block softmin-plus Kogge-Stone, fully inside wave 0
            float y = __shfl(s15, col + 16, 32);   // h_c broadcast to both halves
#pragma unroll
            for (int s = 1; s < 16; s <<= 1) {
                float up = __shfl_up(y, s, 16);
                if ((lane & 15) >= s) y = softmin2f(y, up + 16.0f * (float)s);
            }
            if (lane < 16) carry_l[lane] = y;  // y_c = softmin_{c'<=c}(h_c' + 16(c-c'))
        }
        __syncthreads();                       // (3) scan_s/carry_l visible

        // ---- combine: cur[u] = softmin(intra, carry + j, cur0 + u) ----
        if (tid < Lb) {
            const int c = tid >> 4, j = tid & 15;
            const float sl = scan_s[tid];
            const float ca = (c > 0) ? (carry_l[c - 1] + 1.0f + (float)j) : 1e30f;
            const float zt = cur0 + (float)(tid + 1);
            prevA[tid + 1] = softmin3f(sl, ca, zt);
        }
        if (tid == 0) prevA[0] = cur0;
        // no trailing barrier needed: next-iteration barrier (1) orders
        // prevA writes vs reads, and barriers (2)->(3) cover buffer reuse.
    }

    __syncthreads();
    if (tid == 0) ws[b] = prevA[Lb] / ((float)Lb + 1e-8f);
}

// deterministic serial mean over 64 batch results
__global__ void softper_reduce_kernel(const float* __restrict__ ws,
                                      float* __restrict__ out) {
    if (threadIdx.x == 0) {
        float s = 0.0f;
        for (int i = 0; i < B_DIM; ++i) s += ws[i];
        out[0] = s * (1.0f / (float)B_DIM);
    }
}

extern "C" void kernel_launch(void* const* d_in, const int* in_sizes, int n_in,
                              void* d_out, int out_size, void* d_ws, size_t ws_size,
                              hipStream_t stream) {
    (void)in_sizes; (void)n_in; (void)out_size; (void)ws_size;
    const float* lp      = (const float*)d_in[0];
    const int*   targets = (const int*)d_in[1];
    const int*   ilen    = (const int*)d_in[2];
    const int*   tlen    = (const int*)d_in[3];
    float* ws  = (float*)d_ws;
    float* out = (float*)d_out;

    softper_dp_kernel<<<B_DIM, 256, 0, stream>>>(lp, targets, ilen, tlen, ws);
    softper_reduce_kernel<<<1, 32, 0, stream>>>(ws, out);
}